// MatchLSTM_25013889532628
// MI455X (gfx1250) — compile-verified
//
#include <hip/hip_runtime.h>
#include <math.h>

// ---------------------------------------------------------------------------
// Match-LSTM pointer network forward for MI455X (gfx1250, wave32, WMMA).
// Bulk projections -> parallel f32-WMMA GEMMs; recurrences -> persistent
// single-WGP scan kernels with state in LDS and per-step f32 WMMA tiles.
// All GEMMs are A[M,K] @ Wt[N,K]^T with contiguous v2f (b64) fragment loads;
// right-multiplied weights are pre-transposed once into workspace.
// ---------------------------------------------------------------------------

typedef __attribute__((ext_vector_type(2))) float v2f;
typedef __attribute__((ext_vector_type(8))) float v8f;

#define BB  32
#define TT  400
#define JJ  30
#define DD  150
#define HH  150
#define H2  300
#define H3  450
#define H6  900
#define TP1 401
#define LL  2

__device__ __forceinline__ float sigmoidf_(float x) { return 1.0f / (1.0f + expf(-x)); }

// One 16x16 tile of C = A @ Wt^T.  A: [M,K] row-major (lda), Wt: [N,K] row-major
// (ldw).  M must be a multiple of 16 in this app; K even.  Row/col indices are
// clamped for addressing (garbage lanes only affect output rows/cols that the
// caller never stores), so the K-loop body has NO guards: pure b64 loads + wmma.
// K%4==2 tail: unconditional clamped load + cndmask selects (no divergent mem op).
__device__ __forceinline__ v8f wmma_tile_f32t(const float* A, int lda,
                                              const float* Wt, int ldw,
                                              int K, int mBase, int nBase,
                                              int M, int N) {
  const int lane  = threadIdx.x & 31;
  const int half2 = (lane >> 4) << 1;  // 0 or 2: K sub-offset of this lane half
  const int lr    = lane & 15;
  int arow = mBase + lr; if (arow >= M) arow = M - 1;
  int bcol = nBase + lr; if (bcol >= N) bcol = N - 1;
  const float* Arow = A  + (size_t)arow * lda + half2;
  const float* Wrow = Wt + (size_t)bcol * ldw + half2;
  v8f acc = {0.f, 0.f, 0.f, 0.f, 0.f, 0.f, 0.f, 0.f};
  const int Kf = K & ~3;
#pragma unroll 4
  for (int k = 0; k < Kf; k += 4) {
    const v2f a = *(const v2f*)(Arow + k);
    const v2f b = *(const v2f*)(Wrow + k);
    acc = __builtin_amdgcn_wmma_f32_16x16x4_f32(false, a, false, b, (short)0, acc,
                                                false, false);
  }
  if (Kf < K) {                       // K % 4 == 2 (K = 150 here)
    const bool ok = (Kf + half2 + 1) < K;   // half 0: valid, half 1: beyond K
    const int  kc = ok ? Kf : 0;            // clamped, always-safe address
    const v2f av = *(const v2f*)(Arow + kc);
    const v2f bv = *(const v2f*)(Wrow + kc);
    v2f a, b;
    a.x = ok ? av.x : 0.f;  a.y = ok ? av.y : 0.f;
    b.x = ok ? bv.x : 0.f;  b.y = ok ? bv.y : 0.f;
    acc = __builtin_amdgcn_wmma_f32_16x16x4_f32(false, a, false, b, (short)0, acc,
                                                false, false);
  }
  return acc;
}

// ---------------------------------------------------------------------------
__global__ void embed_kernel(const int* __restrict__ idx, const float* __restrict__ E,
                             float* __restrict__ out, int ntok) {
  int i = blockIdx.x * blockDim.x + threadIdx.x;
  int tok = i / DD, d = i % DD;
  if (tok < ntok) out[i] = E[(size_t)idx[tok] * DD + d];
}

// W[R,C] -> Wt[C,R]  (one-time transpose of right-multiplied weights)
__global__ void transpose_kernel(const float* __restrict__ W, float* __restrict__ Wt,
                                 int R, int C) {
  int i = blockIdx.x * blockDim.x + threadIdx.x;
  if (i >= R * C) return;
  int r = i / C, c = i % C;
  Wt[(size_t)c * R + r] = W[i];
}

// ---------------------------------------------------------------------------
// Bulk WMMA GEMM: C[M,N] = A[M,K] @ Wt[N,K]^T (+ bias[N]).  One wave per tile.
__global__ void gemm_wmma_kernel(const float* __restrict__ A, const float* __restrict__ Wt,
                                 const float* __restrict__ bias, float* __restrict__ C,
                                 int M, int N, int K, int ldw) {
  const int wave = blockIdx.x * (blockDim.x >> 5) + (threadIdx.x >> 5);
  const int nt = (N + 15) >> 4;
  const int mt = (M + 15) >> 4;
  if (wave >= nt * mt) return;  // whole-wave exit: EXEC all-1 for WMMA
  const int mBase = (wave / nt) << 4;
  const int nBase = (wave % nt) << 4;
  v8f acc = wmma_tile_f32t(A, K, Wt, ldw, K, mBase, nBase, M, N);
  const int lane = threadIdx.x & 31;
  const int col = nBase + (lane & 15);
  const int rowOff = (lane >> 4) << 3;
  if (col < N) {
    const float bv = bias ? bias[col] : 0.f;
#pragma unroll
    for (int r = 0; r < 8; ++r) {
      const int row = mBase + rowOff + r;
      if (row < M) C[(size_t)row * N + col] = acc[r] + bv;
    }
  }
}

// ---------------------------------------------------------------------------
// Persistent GRU scan (context & query encoders).
__global__ void __launch_bounds__(512) gru_scan_kernel(
    const float* __restrict__ Gi,   // [B,S,450] = x@Wih^T + bih
    const float* __restrict__ Whh,  // [450,150]
    const float* __restrict__ bhh,  // [450]
    float* __restrict__ Hout,       // [B,S,150]
    int S) {
  extern __shared__ float sm[];
  float* h  = sm;             // [32][160]
  float* gh = sm + 32 * 160;  // [32][456]
  const int tid = threadIdx.x;
  const int wave = tid >> 5;
  for (int i = tid; i < 32 * 160; i += 512) h[i] = 0.f;
  __syncthreads();
  for (int t = 0; t < S; ++t) {
    for (int tile = wave; tile < 58; tile += 16) {          // 2 x 29 tiles
      const int mBase = (tile / 29) << 4;
      const int nBase = (tile % 29) << 4;
      v8f acc = wmma_tile_f32t(h, 160, Whh, HH, HH, mBase, nBase, 32, H3);
      const int lane = tid & 31;
      const int col = nBase + (lane & 15);
      const int rowOff = (lane >> 4) << 3;
      if (col < H3) {
        const float bv = bhh[col];
#pragma unroll
        for (int r = 0; r < 8; ++r) gh[(mBase + rowOff + r) * 456 + col] = acc[r] + bv;
      }
    }
    __syncthreads();
    for (int i = tid; i < 32 * HH; i += 512) {
      const int b = i / HH, c = i % HH;
      const float* gi = Gi + ((size_t)b * S + t) * H3;
      const float r = sigmoidf_(gi[c] + gh[b * 456 + c]);
      const float z = sigmoidf_(gi[HH + c] + gh[b * 456 + HH + c]);
      const float n = tanhf(gi[2 * HH + c] + r * gh[b * 456 + 2 * HH + c]);
      const float hn = (1.f - z) * n + z * h[b * 160 + c];
      h[b * 160 + c] = hn;
      Hout[((size_t)b * S + t) * HH + c] = hn;
    }
    __syncthreads();
  }
}

// ---------------------------------------------------------------------------
// Persistent match-attention scan (one direction per launch).
__global__ void __launch_bounds__(512) match_scan_kernel(
    const float* __restrict__ whq,   // [B,J,150]  Hq@Wq
    const float* __restrict__ Hq,    // [B,J,150]
    const float* __restrict__ WpHp,  // [B,T,150]  Hp@Wp + gate_bias
    const float* __restrict__ GiX,   // [B,T,450]  Hp@Wih_x^T + bih
    const float* __restrict__ WrT,   // [150,150]  transposed: row n = Wr[:,n]
    const float* __restrict__ wvec,  // [150]
    const float* __restrict__ Wih,   // [450,300]  q-columns at +150
    const float* __restrict__ Whh,   // [450,150]
    const float* __restrict__ bhh,   // [450]
    float* __restrict__ Hout,        // [B,T,150]  (scan order)
    int rev) {
  extern __shared__ float sm[];
  float* h  = sm;              // 32*160
  float* gh = h + 32 * 160;    // 32*456
  float* s1 = gh + 32 * 456;   // 32*160
  float* qv = s1 + 32 * 160;   // 32*160
  float* sc = qv + 32 * 160;   // 32*32
  float* av = sc + 32 * 32;    // 32*32
  float* wl = av + 32 * 32;    // 160
  const int tid = threadIdx.x;
  const int wave = tid >> 5;
  for (int i = tid; i < 32 * 160; i += 512) h[i] = 0.f;
  for (int i = tid; i < HH; i += 512) wl[i] = wvec[i];
  __syncthreads();
  for (int t = 0; t < TT; ++t) {
    const int ts = rev ? (TT - 1 - t) : t;
    // pass 1: s1 = h@Wr + WpHp[ts] (tiles 0..19) ; gh = h@Whh^T + bhh (20..77)
    for (int tile = wave; tile < 78; tile += 16) {
      const int lane = tid & 31;
      const int lcol = lane & 15;
      const int rowOff = (lane >> 4) << 3;
      if (tile < 20) {
        const int mBase = (tile / 10) << 4;
        const int nBase = (tile % 10) << 4;
        v8f acc = wmma_tile_f32t(h, 160, WrT, HH, HH, mBase, nBase, 32, HH);
        const int col = nBase + lcol;
        if (col < HH) {
#pragma unroll
          for (int r = 0; r < 8; ++r) {
            const int row = mBase + rowOff + r;
            s1[row * 160 + col] = acc[r] + WpHp[((size_t)row * TT + ts) * HH + col];
          }
        }
      } else {
        const int t2 = tile - 20;
        const int mBase = (t2 / 29) << 4;
        const int nBase = (t2 % 29) << 4;
        v8f acc = wmma_tile_f32t(h, 160, Whh, HH, HH, mBase, nBase, 32, H3);
        const int col = nBase + lcol;
        if (col < H3) {
          const float bv = bhh[col];
#pragma unroll
          for (int r = 0; r < 8; ++r) gh[(mBase + rowOff + r) * 456 + col] = acc[r] + bv;
        }
      }
    }
    __syncthreads();
    // scores[b,j] = sum_h tanh(whq + s1) * w   (attn_bias is softmax-invariant)
    for (int i = tid; i < 32 * JJ; i += 512) {
      const int b = i / JJ, j = i % JJ;
      const float* wq = whq + ((size_t)b * JJ + j) * HH;
      const float* s1b = s1 + b * 160;
      float acc = 0.f;
      for (int c = 0; c < HH; ++c) acc += tanhf(wq[c] + s1b[c]) * wl[c];
      sc[b * 32 + j] = acc;
    }
    __syncthreads();
    if (tid < 32) {  // softmax over J per batch row
      const int b = tid;
      float mx = -1e30f;
      for (int j = 0; j < JJ; ++j) mx = fmaxf(mx, sc[b * 32 + j]);
      float s = 0.f;
      for (int j = 0; j < JJ; ++j) { float e = expf(sc[b * 32 + j] - mx); av[b * 32 + j] = e; s += e; }
      const float inv = 1.f / s;
      for (int j = 0; j < JJ; ++j) av[b * 32 + j] *= inv;
    }
    __syncthreads();
    // attended query qv = a @ Hq
    for (int i = tid; i < 32 * HH; i += 512) {
      const int b = i / HH, c = i % HH;
      const float* hqb = Hq + (size_t)b * JJ * HH + c;
      float acc = 0.f;
      for (int j = 0; j < JJ; ++j) acc += av[b * 32 + j] * hqb[(size_t)j * HH];
      qv[b * 160 + c] = acc;
    }
    __syncthreads();
    // gh += qv @ Wih_q^T   (rows at m_Wih[n*300 + 150 + k])
    for (int tile = wave; tile < 58; tile += 16) {
      const int mBase = (tile / 29) << 4;
      const int nBase = (tile % 29) << 4;
      v8f acc = wmma_tile_f32t(qv, 160, Wih + HH, H2, HH, mBase, nBase, 32, H3);
      const int lane = tid & 31;
      const int col = nBase + (lane & 15);
      const int rowOff = (lane >> 4) << 3;
      if (col < H3) {
#pragma unroll
        for (int r = 0; r < 8; ++r) gh[(mBase + rowOff + r) * 456 + col] += acc[r];
      }
    }
    __syncthreads();
    for (int i = tid; i < 32 * HH; i += 512) {
      const int b = i / HH, c = i % HH;
      const float* gi = GiX + ((size_t)b * TT + ts) * H3;
      const float r = sigmoidf_(gi[c] + gh[b * 456 + c]);
      const float z = sigmoidf_(gi[HH + c] + gh[b * 456 + HH + c]);
      const float n = tanhf(gi[2 * HH + c] + r * gh[b * 456 + 2 * HH + c]);
      const float hn = (1.f - z) * n + z * h[b * 160 + c];
      h[b * 160 + c] = hn;
      Hout[((size_t)b * TT + t) * HH + c] = hn;
    }
    __syncthreads();
  }
}

// ---------------------------------------------------------------------------
// Hr = concat([zeros, concat([Hf, Hb], -1)], axis=1)   -> [B, T+1, 2H]
__global__ void build_hr_kernel(const float* __restrict__ Hf, const float* __restrict__ Hb,
                                float* __restrict__ Hr) {
  const int i = blockIdx.x * blockDim.x + threadIdx.x;
  if (i >= BB * TP1 * H2) return;
  const int c = i % H2;
  const int t = (i / H2) % TP1;
  const int b = i / (H2 * TP1);
  float v = 0.f;
  if (t > 0) {
    const int ts = t - 1;
    v = (c < HH) ? Hf[((size_t)b * TT + ts) * HH + c]
                 : Hb[((size_t)b * TT + ts) * HH + (c - HH)];
  }
  Hr[i] = v;
}

// ---------------------------------------------------------------------------
// Pointer decoder: L=2 attention steps over Hr; writes probs [B,L,T+1].
__global__ void __launch_bounds__(512) ptr_scan_kernel(
    const float* __restrict__ enc,   // [B,401,300] = Hr@ptr_W1
    const float* __restrict__ Hr,    // [B,401,300]
    const float* __restrict__ W2T,   // [300,300] transposed ptr_W2
    const float* __restrict__ vvec,  // [300]
    const float* __restrict__ dWih,  // [900,300]
    const float* __restrict__ dWhh,  // [900,300]
    const float* __restrict__ dbih,  // [900]
    const float* __restrict__ dbhh,  // [900]
    float* __restrict__ out) {       // [32,2,401]
  extern __shared__ float sm[];
  float* h    = sm;                 // 32*304 (persistent)
  float* cbuf = h + 32 * 304;       // 32*304 (attended context)
  float* base = cbuf + 32 * 304;    // phase-overlapped region
  float* hW2  = base;               // 32*304  (phases 1-2)
  float* scb  = base + 32 * 304;    // 32*404  (phases 2-4)
  float* gi   = base;               // 32*912  (phases 5-6; hW2/scb dead)
  float* gh   = base + 32 * 912;    // 32*912
  const int tid = threadIdx.x;
  const int wave = tid >> 5;
  for (int i = tid; i < 32 * 304; i += 512) h[i] = 0.f;
  __syncthreads();
  for (int l = 0; l < LL; ++l) {
    for (int tile = wave; tile < 38; tile += 16) {  // hW2 = h @ W2 : 2x19
      const int mBase = (tile / 19) << 4;
      const int nBase = (tile % 19) << 4;
      v8f acc = wmma_tile_f32t(h, 304, W2T, H2, H2, mBase, nBase, 32, H2);
      const int lane = tid & 31;
      const int col = nBase + (lane & 15);
      const int rowOff = (lane >> 4) << 3;
      if (col < H2) {
#pragma unroll
        for (int r = 0; r < 8; ++r) hW2[(mBase + rowOff + r) * 304 + col] = acc[r];
      }
    }
    __syncthreads();
    for (int i = tid; i < 32 * TP1; i += 512) {  // s = tanh(enc + hW2) . v
      const int b = i / TP1, t = i % TP1;
      const float* eb = enc + ((size_t)b * TP1 + t) * H2;
      const float* hb = hW2 + b * 304;
      float acc = 0.f;
      for (int c = 0; c < H2; ++c) acc += tanhf(eb[c] + hb[c]) * vvec[c];
      scb[b * 404 + t] = acc;
    }
    __syncthreads();
    if (tid < 32) {  // softmax over T+1, write probabilities out
      const int b = tid;
      float mx = -1e30f;
      for (int t = 0; t < TP1; ++t) mx = fmaxf(mx, scb[b * 404 + t]);
      float s = 0.f;
      for (int t = 0; t < TP1; ++t) { float e = expf(scb[b * 404 + t] - mx); scb[b * 404 + t] = e; s += e; }
      const float inv = 1.f / s;
      for (int t = 0; t < TP1; ++t) {
        const float a = scb[b * 404 + t] * inv;
        scb[b * 404 + t] = a;
        out[((size_t)b * LL + l) * TP1 + t] = a;
      }
    }
    __syncthreads();
    for (int i = tid; i < 32 * H2; i += 512) {  // c = a @ Hr
      const int b = i / H2, c = i % H2;
      const float* hrb = Hr + (size_t)b * TP1 * H2 + c;
      float acc = 0.f;
      for (int t = 0; t < TP1; ++t) acc += scb[b * 404 + t] * hrb[(size_t)t * H2];
      cbuf[b * 304 + c] = acc;
    }
    __syncthreads();
    // gi = c@dWih^T + dbih (tiles 0..113), gh = h@dWhh^T + dbhh (114..227)
    for (int tile = wave; tile < 228; tile += 16) {
      const int isGh = tile >= 114;
      const int t2 = isGh ? tile - 114 : tile;
      const int mBase = (t2 / 57) << 4;
      const int nBase = (t2 % 57) << 4;
      const float* Aw = isGh ? h : cbuf;
      const float* Ww = isGh ? dWhh : dWih;
      const float* bw = isGh ? dbhh : dbih;
      float* Ow = isGh ? gh : gi;
      v8f acc = wmma_tile_f32t(Aw, 304, Ww, H2, H2, mBase, nBase, 32, H6);
      const int lane = tid & 31;
      const int col = nBase + (lane & 15);
      const int rowOff = (lane >> 4) << 3;
      if (col < H6) {
        const float bv = bw[col];
#pragma unroll
        for (int r = 0; r < 8; ++r) Ow[(mBase + rowOff + r) * 912 + col] = acc[r] + bv;
      }
    }
    __syncthreads();
    for (int i = tid; i < 32 * H2; i += 512) {  // GRU pointwise, hidden=300
      const int b = i / H2, c = i % H2;
      const float r = sigmoidf_(gi[b * 912 + c] + gh[b * 912 + c]);
      const float z = sigmoidf_(gi[b * 912 + H2 + c] + gh[b * 912 + H2 + c]);
      const float n = tanhf(gi[b * 912 + 2 * H2 + c] + r * gh[b * 912 + 2 * H2 + c]);
      h[b * 304 + c] = (1.f - z) * n + z * h[b * 304 + c];
    }
    __syncthreads();
  }
}

// ---------------------------------------------------------------------------
extern "C" void kernel_launch(void* const* d_in, const int* in_sizes, int n_in,
                              void* d_out, int out_size, void* d_ws, size_t ws_size,
                              hipStream_t stream) {
  const int*   ctx      = (const int*)d_in[0];
  const int*   qry      = (const int*)d_in[1];
  const float* E        = (const float*)d_in[2];
  const float* ctx_Wih  = (const float*)d_in[3];
  const float* ctx_Whh  = (const float*)d_in[4];
  const float* ctx_bih  = (const float*)d_in[5];
  const float* ctx_bhh  = (const float*)d_in[6];
  const float* q_Wih    = (const float*)d_in[7];
  const float* q_Whh    = (const float*)d_in[8];
  const float* q_bih    = (const float*)d_in[9];
  const float* q_bhh    = (const float*)d_in[10];
  const float* Wq       = (const float*)d_in[11];
  const float* Wp       = (const float*)d_in[12];
  const float* Wr       = (const float*)d_in[13];
  const float* wv       = (const float*)d_in[14];
  const float* gateb    = (const float*)d_in[15];
  /* d_in[16] attn_bias: scalar added pre-softmax -> softmax-invariant */
  const float* m_Wih    = (const float*)d_in[17];
  const float* m_Whh    = (const float*)d_in[18];
  const float* m_bih    = (const float*)d_in[19];
  const float* m_bhh    = (const float*)d_in[20];
  const float* mr_Wih   = (const float*)d_in[21];
  const float* mr_Whh   = (const float*)d_in[22];
  const float* mr_bih   = (const float*)d_in[23];
  const float* mr_bhh   = (const float*)d_in[24];
  const float* ptr_W1   = (const float*)d_in[25];
  const float* ptr_W2   = (const float*)d_in[26];
  const float* ptr_v    = (const float*)d_in[27];
  const float* d_Wih    = (const float*)d_in[28];
  const float* d_Whh    = (const float*)d_in[29];
  const float* d_bih    = (const float*)d_in[30];
  const float* d_bhh    = (const float*)d_in[31];
  float* out = (float*)d_out;
  (void)in_sizes; (void)n_in; (void)out_size; (void)ws_size;

  // workspace carve-out (~143 MB of fp32); all offsets stay 8B-aligned
  size_t off = 0;
  auto alloc = [&](size_t n) {
    float* p = (float*)((char*)d_ws + off);
    off += n * sizeof(float);
    return p;
  };
  float* Xc   = alloc((size_t)BB * TT * DD);
  float* Xq   = alloc((size_t)BB * JJ * DD);
  float* GiC  = alloc((size_t)BB * TT * H3);
  float* GiQ  = alloc((size_t)BB * JJ * H3);
  float* Hp   = alloc((size_t)BB * TT * HH);
  float* Hq   = alloc((size_t)BB * JJ * HH);
  float* whqB = alloc((size_t)BB * JJ * HH);
  float* WpHp = alloc((size_t)BB * TT * HH);
  float* GiXf = alloc((size_t)BB * TT * H3);
  float* GiXr = alloc((size_t)BB * TT * H3);
  float* Hf   = alloc((size_t)BB * TT * HH);
  float* Hb   = alloc((size_t)BB * TT * HH);
  float* Hr   = alloc((size_t)BB * TP1 * H2);
  float* encB = alloc((size_t)BB * TP1 * H2);
  float* WqT  = alloc((size_t)HH * HH);
  float* WpT  = alloc((size_t)HH * HH);
  float* WrT  = alloc((size_t)HH * HH);
  float* W1T  = alloc((size_t)H2 * H2);
  float* W2T  = alloc((size_t)H2 * H2);

  auto gemm = [&](const float* A, const float* Wt, const float* bias, float* C,
                  int M, int N, int K, int ldw) {
    const int tiles = ((M + 15) / 16) * ((N + 15) / 16);
    gemm_wmma_kernel<<<(tiles + 7) / 8, 256, 0, stream>>>(A, Wt, bias, C, M, N, K, ldw);
  };

  // 0) one-time transposes of right-multiplied weights (contiguous WMMA B loads)
  transpose_kernel<<<(HH * HH + 255) / 256, 256, 0, stream>>>(Wq, WqT, HH, HH);
  transpose_kernel<<<(HH * HH + 255) / 256, 256, 0, stream>>>(Wp, WpT, HH, HH);
  transpose_kernel<<<(HH * HH + 255) / 256, 256, 0, stream>>>(Wr, WrT, HH, HH);
  transpose_kernel<<<(H2 * H2 + 255) / 256, 256, 0, stream>>>(ptr_W1, W1T, H2, H2);
  transpose_kernel<<<(H2 * H2 + 255) / 256, 256, 0, stream>>>(ptr_W2, W2T, H2, H2);

  // 1) embeddings
  embed_kernel<<<(BB * TT * DD + 255) / 256, 256, 0, stream>>>(ctx, E, Xc, BB * TT);
  embed_kernel<<<(BB * JJ * DD + 255) / 256, 256, 0, stream>>>(qry, E, Xq, BB * JJ);

  // 2) hoisted input projections for both encoder GRUs
  gemm(Xc, ctx_Wih, ctx_bih, GiC, BB * TT, H3, DD, DD);
  gemm(Xq, q_Wih,  q_bih,  GiQ, BB * JJ, H3, DD, DD);

  // 3) sequential encoder scans (persistent single-WGP kernels)
  const size_t ldsGru = (32 * 160 + 32 * 456) * sizeof(float);
  gru_scan_kernel<<<1, 512, ldsGru, stream>>>(GiC, ctx_Whh, ctx_bhh, Hp, TT);
  gru_scan_kernel<<<1, 512, ldsGru, stream>>>(GiQ, q_Whh, q_bhh, Hq, JJ);

  // 4) hoisted match-attention projections
  gemm(Hq, WqT, nullptr, whqB, BB * JJ, HH, HH, HH);          // Hq@Wq
  gemm(Hp, WpT, gateb,  WpHp, BB * TT, HH, HH, HH);           // Hp@Wp + gate_bias
  gemm(Hp, m_Wih,  m_bih,  GiXf, BB * TT, H3, HH, H2);        // x-half of match gates
  gemm(Hp, mr_Wih, mr_bih, GiXr, BB * TT, H3, HH, H2);

  // 5) bidirectional match scans
  const size_t ldsMatch =
      (32 * 160 + 32 * 456 + 32 * 160 + 32 * 160 + 32 * 32 + 32 * 32 + 160) * sizeof(float);
  match_scan_kernel<<<1, 512, ldsMatch, stream>>>(whqB, Hq, WpHp, GiXf, WrT, wv,
                                                  m_Wih, m_Whh, m_bhh, Hf, 0);
  match_scan_kernel<<<1, 512, ldsMatch, stream>>>(whqB, Hq, WpHp, GiXr, WrT, wv,
                                                  mr_Wih, mr_Whh, mr_bhh, Hb, 1);

  // 6) pointer network
  build_hr_kernel<<<(BB * TP1 * H2 + 255) / 256, 256, 0, stream>>>(Hf, Hb, Hr);
  gemm(Hr, W1T, nullptr, encB, BB * TP1, H2, H2, H2);         // enc = Hr@ptr_W1
  const size_t ldsPtr = (size_t)(32 * 304 * 2 + 32 * 912 * 2) * sizeof(float);
  ptr_scan_kernel<<<1, 512, ldsPtr, stream>>>(encB, Hr, W2T, ptr_v,
                                              d_Wih, d_Whh, d_bih, d_bhh, out);
}